// ScoringLayer_20572893348512
// MI455X (gfx1250) — compile-verified
//
#include <hip/hip_runtime.h>
#include <hip/hip_bf16.h>

typedef __attribute__((ext_vector_type(2))) float v2f;
typedef __attribute__((ext_vector_type(4))) float v4f;
typedef __attribute__((ext_vector_type(8))) float v8f;
typedef __attribute__((ext_vector_type(4))) int   v4i;

#define BATCH   64
#define NWIN    256
#define HDIM    768

#define K1_C    0.1f
#define BP_C    1.2f
#define AVDL_C  50.0f
#define ALPHA_C (1.0f / 768.0f)
#define BETA_C  0.1f

// ---------------------------------------------------------------------------
// Kernel A: batched GEMM  C[b] = mask(Q[b]) * mask(D[b])^T  via f32 WMMA.
// Block = 128 threads (4 waves). Block tile = 128x128 of C; wave tile = 64x64.
// K consumed 8 per iteration: one b128 load per fragment covers two WMMA
// k-quads (lanes 0-15 take k0..k0+3, lanes 16-31 take k0+4..k0+7; the
// assignment of k-values to the instruction's logical K slots is free since
// the K-sum is commutative, as long as A and B fragments agree).
// Null-row masks applied at store time (mask zeroes full rows/cols of C).
// ---------------------------------------------------------------------------
__global__ __launch_bounds__(128) void gemm_wmma_f32_kernel(
    const float* __restrict__ q_rep, const float* __restrict__ d_rep,
    const int* __restrict__ q_ids, const int* __restrict__ d_ids,
    float* __restrict__ out_c)
{
    __shared__ float s_mq[128];
    __shared__ float s_md[128];

    const int tid  = threadIdx.x;
    const int lane = tid & 31;
    const int wave = tid >> 5;

    const int blk = blockIdx.x;       // 0..255
    const int b   = blk >> 2;         // batch
    const int t2  = blk & 3;
    const int bm  = (t2 >> 1) * 128;  // block row offset in C
    const int bn  = (t2 & 1) * 128;   // block col offset in C

    // Null-window masks for this block's 128 rows (Q) and 128 cols (D).
    {
        const v4i qi = *(const v4i*)(q_ids + ((size_t)b * NWIN + bm + tid) * 4);
        s_mq[tid] = (qi.x | qi.y | qi.z | qi.w) ? 1.0f : 0.0f;
        const v4i di = *(const v4i*)(d_ids + ((size_t)b * NWIN + bn + tid) * 4);
        s_md[tid] = (di.x | di.y | di.z | di.w) ? 1.0f : 0.0f;
    }
    __syncthreads();

    const int wm = bm + (wave >> 1) * 64;  // wave tile origin (rows)
    const int wn = bn + (wave & 1)  * 64;  // wave tile origin (cols)

    const int lh = lane & 15;              // fragment row/col within 16
    const int kq = (lane >> 4) * 4;        // half-wave k offset within 8-chunk

    const float* qbase = q_rep + (size_t)b * NWIN * HDIM;
    const float* dbase = d_rep + (size_t)b * NWIN * HDIM;

    v8f acc[4][4];
#pragma unroll
    for (int i = 0; i < 4; ++i)
#pragma unroll
        for (int j = 0; j < 4; ++j) acc[i][j] = (v8f){};

    for (int k0 = 0; k0 < HDIM; k0 += 8) {
        v4f af[4], bf[4];
#pragma unroll
        for (int i = 0; i < 4; ++i)
            af[i] = *(const v4f*)(qbase + (size_t)(wm + i * 16 + lh) * HDIM + k0 + kq);
#pragma unroll
        for (int j = 0; j < 4; ++j)
            bf[j] = *(const v4f*)(dbase + (size_t)(wn + j * 16 + lh) * HDIM + k0 + kq);

#pragma unroll
        for (int i = 0; i < 4; ++i) {
            v2f a0; a0.x = af[i].x; a0.y = af[i].y;
            v2f a1; a1.x = af[i].z; a1.y = af[i].w;
#pragma unroll
            for (int j = 0; j < 4; ++j) {
                v2f b0; b0.x = bf[j].x; b0.y = bf[j].y;
                v2f b1; b1.x = bf[j].z; b1.y = bf[j].w;
                acc[i][j] = __builtin_amdgcn_wmma_f32_16x16x4_f32(
                    false, a0, false, b0, (short)0, acc[i][j], false, false);
                acc[i][j] = __builtin_amdgcn_wmma_f32_16x16x4_f32(
                    false, a1, false, b1, (short)0, acc[i][j], false, false);
            }
        }
    }

    // Store: C/D layout = reg r, lane l -> row r + 8*(l>=16), col l&15.
    const int rowLds = (wave >> 1) * 64;   // wave row offset inside block LDS masks
    const int colLds = (wave & 1)  * 64;
    const int rhalf  = (lane >> 4) * 8;

#pragma unroll
    for (int i = 0; i < 4; ++i) {
        const int rsub = i * 16 + rhalf;   // row offset (within wave tile) of reg 0
        float mq[8];
#pragma unroll
        for (int r = 0; r < 8; ++r) mq[r] = s_mq[rowLds + rsub + r];
#pragma unroll
        for (int j = 0; j < 4; ++j) {
            const float md  = s_md[colLds + j * 16 + lh];
            const int   col = wn + j * 16 + lh;
            float* crow = out_c + (size_t)b * NWIN * NWIN
                                + (size_t)(wm + rsub) * NWIN + col;
#pragma unroll
            for (int r = 0; r < 8; ++r)
                crow[(size_t)r * NWIN] = acc[i][j][r] * mq[r] * md;
        }
    }
}

// ---------------------------------------------------------------------------
// Kernel B: per-(b,q) BM25 scoring. One 256-thread block per (b,q); thread t
// handles document window d = t. Coalesced C-row read, int4 exact-match,
// LDS tree reduction of (tf * d_tfs) and of dl = sum(d_tfs).
// ---------------------------------------------------------------------------
__global__ __launch_bounds__(256) void score_kernel(
    const float* __restrict__ c_mat,   // [B, NW, NW] (d_out + 64)
    const float* __restrict__ qtw,     // [B, NW]
    const int* __restrict__ q_ids,     // [B, NW, 4]
    const int* __restrict__ d_ids,     // [B, NW, 4]
    const int* __restrict__ d_tfs,     // [B, NW]
    float* __restrict__ ws)            // [B*NW] per-q contributions
{
    const int bq = blockIdx.x;         // 0..B*NW-1
    const int b  = bq >> 8;
    const int qw = bq & 255;
    const int t  = threadIdx.x;

    const v4i qi = *(const v4i*)(q_ids + ((size_t)b * NWIN + qw) * 4);
    const v4i di = *(const v4i*)(d_ids + ((size_t)b * NWIN + t) * 4);
    const float em = (qi.x == di.x && qi.y == di.y &&
                      qi.z == di.z && qi.w == di.w) ? 1.0f : 0.0f;

    const float c   = c_mat[(size_t)b * NWIN * NWIN + (size_t)qw * NWIN + t];
    const float tfm = (float)d_tfs[(size_t)b * NWIN + t];
    const float val = (c * (ALPHA_C * BETA_C) + em * (1.0f - BETA_C)) * tfm;

    __shared__ float red1[256];
    __shared__ float red2[256];
    red1[t] = val;
    red2[t] = tfm;
    __syncthreads();
#pragma unroll
    for (int s = 128; s > 0; s >>= 1) {
        if (t < s) { red1[t] += red1[t + s]; red2[t] += red2[t + s]; }
        __syncthreads();
    }
    if (t == 0) {
        const float etdf  = red1[0];
        const float dl    = red2[0];
        const float denom = etdf * (1.0f + K1_C);
        const float nom   = etdf + K1_C * (1.0f - BP_C + BP_C * dl / AVDL_C);
        const float dtw   = denom / (nom + 1e-8f);
        ws[bq] = qtw[(size_t)b * NWIN + qw] * dtw;
    }
}

// ---------------------------------------------------------------------------
// Kernel C: reduce 256 per-q contributions into s[b].
// ---------------------------------------------------------------------------
__global__ __launch_bounds__(256) void final_reduce_kernel(
    const float* __restrict__ ws, float* __restrict__ s_out)
{
    const int b = blockIdx.x;
    const int t = threadIdx.x;
    __shared__ float red[256];
    red[t] = ws[(size_t)b * NWIN + t];
    __syncthreads();
#pragma unroll
    for (int s = 128; s > 0; s >>= 1) {
        if (t < s) red[t] += red[t + s];
        __syncthreads();
    }
    if (t == 0) s_out[b] = red[0];
}

// ---------------------------------------------------------------------------
extern "C" void kernel_launch(void* const* d_in, const int* in_sizes, int n_in,
                              void* d_out, int out_size, void* d_ws, size_t ws_size,
                              hipStream_t stream) {
    (void)in_sizes; (void)n_in; (void)out_size; (void)ws_size;

    const float* q_rep = (const float*)d_in[0];
    const float* d_rep = (const float*)d_in[1];
    const float* qtw   = (const float*)d_in[2];
    const int*   q_ids = (const int*)d_in[3];
    const int*   d_ids = (const int*)d_in[4];
    const int*   d_tfs = (const int*)d_in[5];

    float* out   = (float*)d_out;
    float* s_out = out;        // [64]
    float* c_out = out + 64;   // [64, 256, 256] d_expanded_tf
    float* ws    = (float*)d_ws;  // [64*256] floats

    gemm_wmma_f32_kernel<<<BATCH * 4, 128, 0, stream>>>(q_rep, d_rep, q_ids, d_ids, c_out);
    score_kernel<<<BATCH * NWIN, 256, 0, stream>>>(c_out, qtw, q_ids, d_ids, d_tfs, ws);
    final_reduce_kernel<<<BATCH, 256, 0, stream>>>(ws, s_out);
}